// LightGCN_44367012168079
// MI455X (gfx1250) — compile-verified
//
#include <hip/hip_runtime.h>

// LightGCN propagation for MI455X (gfx1250).
//
// Roofline: 3 layers x 5M edges x 64 dims x 2 FLOP ~= 1.9 GFLOP vs ~GBs of
// gather/scatter traffic -> ~0.25 FLOP/byte. WMMA is inapplicable (sparse
// random adjacency); the chip features that matter are:
//  - 192MB L2: the 38.4MB feature table + 38.4MB accumulator stay L2-resident,
//    so gathers and scatter-atomics are L2 traffic, not HBM traffic.
//  - native f32 atomics at L2 (global_atomic_add_f32 via inline asm, never CAS)
//  - Tensor Data Mover: DMA the edge stream into LDS, double-buffered
//    (TENSORcnt + s_wait_tensorcnt), removing 16x-redundant VMEM index loads.
//  - b128 gathers: 16 lanes x float4 = one 64-dim node row per edge.

#define DIM 64
#define EDGES_PER_CHUNK 256
#define CHUNKS_PER_BLOCK 8
#define EDGES_PER_BLOCK (EDGES_PER_CHUNK * CHUNKS_PER_BLOCK)

typedef unsigned int u32x4 __attribute__((ext_vector_type(4)));
typedef unsigned int u32x8 __attribute__((ext_vector_type(8)));

// ---- CDNA5 native f32 atomic-add, device scope, fire-and-forget ------------
__device__ __forceinline__ void gatomic_add_f32(float* p, float v) {
    asm volatile("global_atomic_add_f32 %0, %1, off scope:SCOPE_DEV"
                 :: "v"(p), "v"(v)
                 : "memory");
}

// ---- TDM: DMA a 2-row tile (row[],col[] chunk) from global into LDS --------
// 2D tile: tile_dim0 = EDGES_PER_CHUNK elements, tile_dim1 = 2 rows,
// rows separated by tensor_dim0_stride = E elements. data_size = 4 bytes.
// tensor_dim0 = remaining elements so tail chunks OOB-read as zeros.
__device__ __forceinline__ void tdm_load_edges(unsigned lds_off, const int* gptr,
                                               unsigned remaining,
                                               unsigned stride_elems) {
    unsigned long long ga = (unsigned long long)(uintptr_t)gptr;
    u32x4 g0;
    g0.x = 0x1u;                                   // count=1, user descriptor
    g0.y = lds_off;                                // LDS byte address
    g0.z = (unsigned)(ga & 0xffffffffu);           // global_addr[31:0]
    g0.w = (unsigned)((ga >> 32) & 0x01ffffffu)    // global_addr[56:32]
         | (2u << 30);                             // type = 2 ("image")
    u32x8 g1;
    g1.s0 = (2u << 16);                            // data_size = 4B (code 2)
    g1.s1 = (remaining & 0xffffu) << 16;           // tensor_dim0[15:0]
    g1.s2 = ((remaining >> 16) & 0xffffu)          // tensor_dim0[31:16]
          | (2u << 16);                            // tensor_dim1 = 2
    g1.s3 = ((unsigned)EDGES_PER_CHUNK << 16);     // tile_dim0
    g1.s4 = 2u;                                    // tile_dim1 = 2 (tile_dim2=0)
    g1.s5 = stride_elems;                          // tensor_dim0_stride[31:0]
    g1.s6 = 0u;                                    // stride hi / dim1_stride lo
    g1.s7 = 0u;
    asm volatile("tensor_load_to_lds %0, %1" :: "s"(g0), "s"(g1) : "memory");
}

// ---- zero a float4 region (grid-stride) ------------------------------------
__global__ __launch_bounds__(256) void zero_f4_kernel(float4* __restrict__ p, long n4) {
    long i      = (long)blockIdx.x * blockDim.x + threadIdx.x;
    long stride = (long)gridDim.x * blockDim.x;
    const float4 z = make_float4(0.f, 0.f, 0.f, 0.f);
    for (; i < n4; i += stride) p[i] = z;
}

// ---- build x = concat(user_emb, item_emb); seed acc = 0.25 * x -------------
__global__ __launch_bounds__(256) void init_concat_kernel(
        const float4* __restrict__ user_emb, const float4* __restrict__ item_emb,
        long u4, long n4, float4* __restrict__ x, float4* __restrict__ acc) {
    long i = (long)blockIdx.x * blockDim.x + threadIdx.x;
    if (i >= n4) return;
    float4 v = (i < u4) ? user_emb[i] : item_emb[i - u4];
    x[i] = v;
    acc[i] = make_float4(0.25f * v.x, 0.25f * v.y, 0.25f * v.z, 0.25f * v.w);
}

// ---- degree count at targets: deg[col[e]] += 1.0 ---------------------------
__global__ __launch_bounds__(256) void degree_kernel(
        const int* __restrict__ col, long E, float* __restrict__ deg) {
    long e = (long)blockIdx.x * blockDim.x + threadIdx.x;
    if (e >= E) return;
    int c = __builtin_nontemporal_load(col + e);   // read-once stream -> TH_NT
    gatomic_add_f32(deg + c, 1.0f);
}

// ---- deg -> deg^{-1/2} in place (0 stays 0) --------------------------------
__global__ __launch_bounds__(256) void rsqrt_kernel(float* __restrict__ deg, long n) {
    long i = (long)blockIdx.x * blockDim.x + threadIdx.x;
    if (i >= n) return;
    float d = deg[i];
    deg[i] = (d > 0.f) ? __frsqrt_rn(d) : 0.f;
}

// ---- one LGConv layer: xn[col] += (dis[row]*dis[col]) * x[row] -------------
// Block = 256 threads = 16 groups of 16 lanes. Edge index chunks are DMA'd
// into LDS by the TDM (double-buffered, wave 0 issues, TENSORcnt pipelined);
// each group then walks 16 edges per chunk reading indices via LDS broadcast,
// gathers the 64-dim node row as 16 x float4 (b128, L2-resident), and
// scatter-adds with native f32 atomics performed at L2.
__global__ __launch_bounds__(256) void propagate_kernel(
        const float* __restrict__ x, float* __restrict__ xn,
        const int* __restrict__ edges,   // row = edges[0..E), col = edges[E..2E)
        const float* __restrict__ dis, long E) {
    __shared__ int ebuf[2][2 * EDGES_PER_CHUNK];   // [buf][row(256) | col(256)]

    const int lane = threadIdx.x & 15;   // dim group: floats [4*lane, 4*lane+4)
    const int grp  = threadIdx.x >> 4;   // edge slot within chunk: 0..15
    const bool issuer = (threadIdx.x < 32);        // wave 0 drives the TDM

    const long base = (long)blockIdx.x * EDGES_PER_BLOCK;
    const long rem_block = E - base;
    if (rem_block <= 0) return;
    const int nchunks = (int)((rem_block < (long)EDGES_PER_BLOCK
                                  ? rem_block : (long)EDGES_PER_BLOCK)
                              + EDGES_PER_CHUNK - 1) / EDGES_PER_CHUNK;

    const unsigned lds0 = (unsigned)(uintptr_t)(&ebuf[0][0]);
    const unsigned lds1 = (unsigned)(uintptr_t)(&ebuf[1][0]);

    if (issuer) {
        tdm_load_edges(lds0, edges + base, (unsigned)(E - base), (unsigned)E);
    }

    for (int k = 0; k < nchunks; ++k) {
        if (issuer) {
            if (k + 1 < nchunks) {
                long nb = base + (long)(k + 1) * EDGES_PER_CHUNK;
                tdm_load_edges((k & 1) ? lds0 : lds1, edges + nb,
                               (unsigned)(E - nb), (unsigned)E);
                __builtin_amdgcn_s_wait_tensorcnt(1);  // chunk k landed
            } else {
                __builtin_amdgcn_s_wait_tensorcnt(0);  // last chunk landed
            }
        }
        __syncthreads();                                // data visible to block

        const int* lrow = &ebuf[k & 1][0];
        const int* lcol = &ebuf[k & 1][EDGES_PER_CHUNK];
        const long ebase = base + (long)k * EDGES_PER_CHUNK;
#pragma unroll 4
        for (int j = 0; j < EDGES_PER_CHUNK / 16; ++j) {
            const int idx = grp * (EDGES_PER_CHUNK / 16) + j;
            const long e = ebase + idx;
            if (e >= E) break;                          // uniform within group
            int r = lrow[idx];                          // LDS broadcast read
            int c = lcol[idx];
            float w = dis[r] * dis[c];                  // 0.6MB table: cache-hit
            float4 v = *((const float4*)(x + (long)r * DIM) + lane);  // b128
            float* dst = xn + (long)c * DIM + (long)lane * 4;
            gatomic_add_f32(dst + 0, v.x * w);
            gatomic_add_f32(dst + 1, v.y * w);
            gatomic_add_f32(dst + 2, v.z * w);
            gatomic_add_f32(dst + 3, v.w * w);
        }
        __syncthreads();                                // buffer safe to reuse
    }
}

// ---- acc += 0.25 * xn; zero the retiring buffer for reuse as next xn -------
__global__ __launch_bounds__(256) void add_clear_kernel(
        float4* __restrict__ acc, const float4* __restrict__ xn,
        float4* __restrict__ x_old, long n4) {
    long i = (long)blockIdx.x * blockDim.x + threadIdx.x;
    if (i >= n4) return;
    float4 v = xn[i];
    float4 a = acc[i];
    a.x += 0.25f * v.x; a.y += 0.25f * v.y;
    a.z += 0.25f * v.z; a.w += 0.25f * v.w;
    acc[i]   = a;
    x_old[i] = make_float4(0.f, 0.f, 0.f, 0.f);
}

static inline long align256(long b) { return (b + 255) & ~255L; }

extern "C" void kernel_launch(void* const* d_in, const int* in_sizes, int n_in,
                              void* d_out, int out_size, void* d_ws, size_t ws_size,
                              hipStream_t stream) {
    const float* user_emb = (const float*)d_in[0];
    const float* item_emb = (const float*)d_in[1];
    const int*   edges    = (const int*)d_in[2];   // [2, E] int32
    float*       out      = (float*)d_out;         // [N, 64] = acc / 4

    const long U  = in_sizes[0];        // user floats  (100000*64)
    const long I  = in_sizes[1];        // item floats  (50000*64)
    const long ND = U + I;              // total node floats
    const long N  = ND / DIM;           // nodes
    const long E  = (long)in_sizes[2] / 2;
    const int* col = edges + E;

    // Workspace: dis[N] | xa[ND] | xb[ND]
    const long dis_bytes = align256(N * 4);
    const long x_bytes   = align256(ND * 4);
    if ((long)ws_size < dis_bytes + 2 * x_bytes) return;  // can't run
    char* w = (char*)d_ws;
    float* dis = (float*)w;                 w += dis_bytes;
    float* xa  = (float*)w;                 w += x_bytes;
    float* xb  = (float*)w;

    const int TPB = 256;
    const long n4 = ND / 4;             // DIM=64 -> always divisible
    const long d4 = N / 4;
    const int grid_n4   = (int)((n4 + TPB - 1) / TPB);
    const int grid_d4   = (int)((d4 + TPB - 1) / TPB);
    const int grid_n    = (int)((N + TPB - 1) / TPB);
    const int grid_E    = (int)((E + TPB - 1) / TPB);
    const int grid_prop = (int)((E + EDGES_PER_BLOCK - 1) / EDGES_PER_BLOCK);

    // 1) clear degree accumulator and first xnext buffer
    zero_f4_kernel<<<grid_d4, TPB, 0, stream>>>((float4*)dis, d4);
    zero_f4_kernel<<<grid_n4, TPB, 0, stream>>>((float4*)xb, n4);

    // 2) x = concat(user, item); acc = 0.25 * x  (layer-0 term folded in)
    init_concat_kernel<<<grid_n4, TPB, 0, stream>>>(
        (const float4*)user_emb, (const float4*)item_emb,
        U / 4, n4, (float4*)xa, (float4*)out);

    // 3) degrees at targets, then deg^{-1/2}
    degree_kernel<<<grid_E, TPB, 0, stream>>>(col, E, dis);
    rsqrt_kernel<<<grid_n, TPB, 0, stream>>>(dis, N);

    // 4) three LGConv layers, ping-ponging xa/xb
    float* xcur = xa;
    float* xnxt = xb;
    for (int layer = 0; layer < 3; ++layer) {
        propagate_kernel<<<grid_prop, TPB, 0, stream>>>(xcur, xnxt, edges, dis, E);
        // acc += 0.25*xnxt; clear xcur so it can serve as the next xnext
        add_clear_kernel<<<grid_n4, TPB, 0, stream>>>(
            (float4*)out, (const float4*)xnxt, (float4*)xcur, n4);
        float* t = xcur; xcur = xnxt; xnxt = t;
    }
    // d_out already holds (x0+x1+x2+x3)/4 laid out as [users; items].
}